// Net_64939905515613
// MI455X (gfx1250) — compile-verified
//
#include <hip/hip_runtime.h>
#include <hip/hip_bf16.h>
#include <math.h>

typedef __attribute__((ext_vector_type(2))) float v2f;
typedef __attribute__((ext_vector_type(8))) float v8f;

#define NNODES 100000
#define NEDGES 1600000

// ---------------------------------------------------------------------------
// Pack W1[2][32][64] and root1[32][64] into Wcat1[32][192] = [W1_0 | W1_1 | root1]
// ---------------------------------------------------------------------------
__global__ void pack_w1_kernel(const float* __restrict__ W1,
                               const float* __restrict__ root1,
                               float* __restrict__ Wcat1) {
    int t = blockIdx.x * blockDim.x + threadIdx.x;
    if (t >= 32 * 192) return;
    int f = t / 192;
    int c = t % 192;
    float v;
    if (c < 64)        v = W1[f * 64 + c];               // k=0
    else if (c < 128)  v = W1[2048 + f * 64 + (c - 64)]; // k=1
    else               v = root1[f * 64 + (c - 128)];    // root
    Wcat1[t] = v;
}

// ---------------------------------------------------------------------------
// Pack W2[2][64][40] and root2[64][40] into Wcat2[64][128] (zero-padded 120..127)
// ---------------------------------------------------------------------------
__global__ void pack_w2_kernel(const float* __restrict__ W2,
                               const float* __restrict__ root2,
                               float* __restrict__ Wcat2) {
    int t = blockIdx.x * blockDim.x + threadIdx.x;
    if (t >= 64 * 128) return;
    int f = t / 128;
    int c = t % 128;
    float v;
    if (c < 40)        v = W2[f * 40 + c];                // k=0
    else if (c < 80)   v = W2[2560 + f * 40 + (c - 40)];  // k=1
    else if (c < 120)  v = root2[f * 40 + (c - 80)];      // root
    else               v = 0.0f;                          // pad
    Wcat2[t] = v;
}

// ---------------------------------------------------------------------------
// FP32 WMMA row-strip GEMM: D[M x NT*16] = A[M x KDIM] * B[KDIM x NT*16].
// One wave32 owns a 16-row strip and ALL NT column tiles: A is loaded exactly
// once per K-step (8-byte load per lane) and each K-step issues NT independent
// V_WMMA_F32_16X16X4_F32 ops into separate accumulators (no RAW hazards,
// maximal XDL-pipe occupancy). B (<=32KB) stays cache-resident.
//
// ISA layouts (CDNA5 ISA):
//   A 16x4  f32: lanes 0-15 -> M=lane, VGPR0=K0, VGPR1=K1; lanes 16-31 -> K2,K3
//   B 4x16  f32: lanes 0-15 -> N=lane, VGPR0=K0, VGPR1=K1; lanes 16-31 -> K2,K3
//   C/D 16x16 f32: VGPR r, lanes 0-15 -> M=r, N=lane; lanes 16-31 -> M=8+r
// ---------------------------------------------------------------------------
template <int KDIM, int NT>
__global__ __launch_bounds__(128) void gemm_wmma_rowstrip_kernel(
    const float* __restrict__ A,
    const float* __restrict__ B,
    float* __restrict__ D,
    int M) {
    constexpr int Nc = NT * 16;
    const int lane  = threadIdx.x & 31;
    const int gwave = blockIdx.x * (blockDim.x >> 5) + (threadIdx.x >> 5);
    if (gwave >= (M >> 4)) return;          // wave-uniform exit (EXEC all-1 for WMMA)

    const int lmod  = lane & 15;
    const int row   = gwave * 16 + lmod;    // A row for this lane
    const int khalf = (lane >> 4) << 1;     // lanes 0-15: +0, lanes 16-31: +2

    v8f acc[NT];
#pragma unroll
    for (int t = 0; t < NT; ++t) acc[t] = (v8f){};

#pragma unroll
    for (int kk = 0; kk < KDIM; kk += 4) {
        const int k0 = kk + khalf;
        // A fragment: two consecutive K values -> one 8-byte load, reused NT times
        v2f a = *(const v2f*)(A + (size_t)row * KDIM + k0);
#pragma unroll
        for (int t = 0; t < NT; ++t) {
            const int col = t * 16 + lmod;
            v2f b;
            b.x = B[(size_t)k0 * Nc + col];
            b.y = B[(size_t)(k0 + 1) * Nc + col];
            acc[t] = __builtin_amdgcn_wmma_f32_16x16x4_f32(
                /*neg_a=*/false, a, /*neg_b=*/false, b,
                /*c_mod=*/(short)0, acc[t], /*reuse_a=*/false, /*reuse_b=*/false);
        }
    }

    const int rbase = gwave * 16 + ((lane >> 4) << 3);
#pragma unroll
    for (int t = 0; t < NT; ++t) {
        const int col = t * 16 + lmod;
#pragma unroll
        for (int r = 0; r < 8; ++r) {
            D[(size_t)(rbase + r) * Nc + col] = acc[t][r];
        }
    }
}

// ---------------------------------------------------------------------------
// Edge kernel: one wave32 per edge. Linear spline interpolation (K=2) of two
// gathered xW rows + float-atomic scatter into agg[dst]. Each lane owns two
// contiguous channels -> one global_load_b64 per gathered row per lane.
// C = channels per spline slot; STRIDE = xw row stride.
// ---------------------------------------------------------------------------
template <int C, int STRIDE, int ADDCNT>
__global__ __launch_bounds__(256) void edge_msg_kernel(
    const int* __restrict__ src, const int* __restrict__ dst,
    const float* __restrict__ pseudo,
    const float* __restrict__ xw,
    float* __restrict__ agg, float* __restrict__ cnt,
    int E) {
    const int lane = threadIdx.x & 31;
    const int e = blockIdx.x * (blockDim.x >> 5) + (threadIdx.x >> 5);
    if (e >= E) return;

    const int s = src[e];
    const int d = dst[e];
    const float p  = pseudo[e];           // v = p*(K-1) = p for K=2
    const float lo = floorf(p);
    const float frac = p - lo;
    int i0 = (int)lo;
    int i1 = i0 + 1; if (i1 > 1) i1 = 1;  // min(lo+1, K-1)
    const float w0 = 1.0f - frac;
    const float w1 = frac;

    const float* r0 = xw + (size_t)s * STRIDE + (size_t)i0 * C;
    const float* r1 = xw + (size_t)s * STRIDE + (size_t)i1 * C;
    float* ag = agg + (size_t)d * C;

#pragma unroll
    for (int c0 = 2 * lane; c0 < C; c0 += 64) {   // C=64: all lanes; C=40: lanes 0-19
        v2f a = *(const v2f*)(r0 + c0);
        v2f b = *(const v2f*)(r1 + c0);
        float m0 = w0 * a.x + w1 * b.x;
        float m1 = w0 * a.y + w1 * b.y;
        atomicAdd(&ag[c0],     m0);
        atomicAdd(&ag[c0 + 1], m1);
    }
    if (ADDCNT && lane == 0) atomicAdd(&cnt[d], 1.0f);
}

// ---------------------------------------------------------------------------
// Layer-1 epilogue: h = elu(agg/max(cnt,1) + root_part + b1)
// root part lives in xw1 columns [128,192).
// ---------------------------------------------------------------------------
__global__ __launch_bounds__(256) void finalize1_kernel(
    const float* __restrict__ agg, const float* __restrict__ cnt,
    const float* __restrict__ xw1, const float* __restrict__ b1,
    float* __restrict__ h, int Nn) {
    int t = blockIdx.x * blockDim.x + threadIdx.x;
    if (t >= Nn * 64) return;
    int n = t >> 6;
    int c = t & 63;
    float cn = fmaxf(cnt[n], 1.0f);
    float v = agg[t] / cn + xw1[(size_t)n * 192 + 128 + c] + b1[c];
    h[t] = (v > 0.0f) ? v : expm1f(v);    // elu, alpha=1
}

// ---------------------------------------------------------------------------
// Layer-2 epilogue + log_softmax: one wave32 per node (40 outputs).
// root part lives in xw2 columns [80,120) (row stride 128).
// ---------------------------------------------------------------------------
__global__ __launch_bounds__(256) void finalize2_kernel(
    const float* __restrict__ agg, const float* __restrict__ cnt,
    const float* __restrict__ xw2, const float* __restrict__ b2,
    float* __restrict__ out, int Nn) {
    const int lane = threadIdx.x & 31;
    const int n = blockIdx.x * (blockDim.x >> 5) + (threadIdx.x >> 5);
    if (n >= Nn) return;

    float cn = fmaxf(cnt[n], 1.0f);
    const int c0 = lane;
    const int c1 = lane + 32;
    const size_t arow = (size_t)n * 40;
    const size_t xrow = (size_t)n * 128 + 80;

    float v0 = agg[arow + c0] / cn + xw2[xrow + c0] + b2[c0];
    float v1 = -INFINITY;
    if (c1 < 40) v1 = agg[arow + c1] / cn + xw2[xrow + c1] + b2[c1];

    float m = fmaxf(v0, v1);
#pragma unroll
    for (int off = 16; off > 0; off >>= 1)
        m = fmaxf(m, __shfl_xor(m, off, 32));

    float s = __expf(v0 - m) + ((c1 < 40) ? __expf(v1 - m) : 0.0f);
#pragma unroll
    for (int off = 16; off > 0; off >>= 1)
        s += __shfl_xor(s, off, 32);

    float lse = m + __logf(s);
    out[arow + c0] = v0 - lse;
    if (c1 < 40) out[arow + c1] = v1 - lse;
}

// ---------------------------------------------------------------------------
extern "C" void kernel_launch(void* const* d_in, const int* in_sizes, int n_in,
                              void* d_out, int out_size, void* d_ws, size_t ws_size,
                              hipStream_t stream) {
    const float* x     = (const float*)d_in[0];
    const int*   eidx  = (const int*)  d_in[1];
    const float* eattr = (const float*)d_in[2];
    const float* W1    = (const float*)d_in[3];
    const float* root1 = (const float*)d_in[4];
    const float* b1    = (const float*)d_in[5];
    const float* W2    = (const float*)d_in[6];
    const float* root2 = (const float*)d_in[7];
    const float* b2    = (const float*)d_in[8];
    float* out = (float*)d_out;

    const int Nn = NNODES;
    const int E  = NEDGES;
    const int* src = eidx;
    const int* dst = eidx + E;

    // -------- workspace carve-out (floats), 64-elem aligned --------
    float* ws = (float*)d_ws;
    size_t off = 0;
    auto take = [&](size_t nfloats) {
        float* p = ws + off;
        off += (nfloats + 63) & ~(size_t)63;
        return p;
    };
    float* cnt   = take(Nn);                 // in-degree counts
    float* Wcat1 = take(32 * 192);           // packed layer-1 weights
    float* Wcat2 = take(64 * 128);           // packed layer-2 weights (padded)
    float* xw    = take((size_t)Nn * 192);   // xW1; reused as xW2 (stride 128)
    float* agg   = take((size_t)Nn * 64);    // agg1; reused as agg2 (40 cols)
    float* h     = take((size_t)Nn * 64);    // layer-1 activations

    // -------- zero accumulators (fresh every call; graph-capture safe) --------
    hipMemsetAsync(cnt, 0, (size_t)Nn * sizeof(float), stream);
    hipMemsetAsync(agg, 0, (size_t)Nn * 64 * sizeof(float), stream);

    // -------- pack weights --------
    pack_w1_kernel<<<(32 * 192 + 255) / 256, 256, 0, stream>>>(W1, root1, Wcat1);
    pack_w2_kernel<<<(64 * 128 + 255) / 256, 256, 0, stream>>>(W2, root2, Wcat2);

    // -------- layer 1: xw = x[N,32] @ Wcat1[32,192] --------
    {
        int waves  = Nn / 16;                // one wave per 16-row strip
        int blocks = (waves + 3) / 4;        // 128 threads = 4 waves/block
        gemm_wmma_rowstrip_kernel<32, 12><<<blocks, 128, 0, stream>>>(x, Wcat1, xw, Nn);
    }
    edge_msg_kernel<64, 192, 1><<<(E + 7) / 8, 256, 0, stream>>>(src, dst, eattr,
                                                                 xw, agg, cnt, E);
    finalize1_kernel<<<(Nn * 64 + 255) / 256, 256, 0, stream>>>(agg, cnt, xw, b1, h, Nn);

    // -------- layer 2: xw = h[N,64] @ Wcat2[64,128] --------
    hipMemsetAsync(agg, 0, (size_t)Nn * 64 * sizeof(float), stream);
    {
        int waves  = Nn / 16;
        int blocks = (waves + 3) / 4;
        gemm_wmma_rowstrip_kernel<64, 8><<<blocks, 128, 0, stream>>>(h, Wcat2, xw, Nn);
    }
    edge_msg_kernel<40, 128, 0><<<(E + 7) / 8, 256, 0, stream>>>(src, dst, eattr,
                                                                 xw, agg, cnt, E);
    finalize2_kernel<<<(Nn + 7) / 8, 256, 0, stream>>>(agg, cnt, xw, b2, out, Nn);
}